// PytorchLlamaKVUpdateModel_81063212745031
// MI455X (gfx1250) — compile-verified
//
#include <hip/hip_runtime.h>
#include <hip/hip_bf16.h>

// ---------------------------------------------------------------------------
// KV-cache update for MI455X (gfx1250).
// Pure streaming op (~520 MiB moved, 0 FLOPs) -> HBM-bound (~23 us @ 23.3 TB/s).
// Bulk copy uses the CDNA5 async global<->LDS data-mover path
// (GLOBAL_LOAD_ASYNC_TO_LDS_B128 / GLOBAL_STORE_ASYNC_FROM_LDS_B128, ASYNCcnt)
// with TH=NT cache policy (stream is 2.7x the 192 MB L2, never re-read).
// The transpose-scatter uses NT B128 loads/stores.
// ---------------------------------------------------------------------------

#define SEQLEN     128
#define N_KV_HEADS 8
#define BATCH      8
#define HEAD_DIM   128
#define MAX_SEQ    4096

// 16-byte element counts
#define PER_T4   ((size_t)BATCH * N_KV_HEADS * MAX_SEQ * (HEAD_DIM / 4))   // 8,388,608 per cache tensor
#define X_T4     ((size_t)SEQLEN * N_KV_HEADS * BATCH * (HEAD_DIM / 4))    // 262,144 per new tensor
#define TOTAL_T4 (2 * PER_T4)                                              // 16,777,216

#define BLK   256
#define CHUNK 4                 // 16B packets per thread per tile
#define TILE  (BLK * CHUNK)     // 1024 packets = 16 KiB per block-tile

// CPol for gfx1250 VMEM: TH in bits [2:0]; TH=1 => non-temporal
#define CPOL_NT 1

// clang-native 16B vector types (NT builtins & async builtins need these)
typedef float v4f  __attribute__((ext_vector_type(4)));
typedef int   b128_t __attribute__((vector_size(16)));   // matches builtin param "V4i"

#if defined(__HIP_DEVICE_COMPILE__) && defined(__gfx1250__) && \
    __has_builtin(__builtin_amdgcn_global_load_async_to_lds_b128) && \
    __has_builtin(__builtin_amdgcn_global_store_async_from_lds_b128) && \
    __has_builtin(__builtin_amdgcn_s_wait_asynccnt)
#define HAVE_ASYNC_LDS 1
typedef __attribute__((address_space(1))) b128_t* glb_b128_ptr;
typedef __attribute__((address_space(3))) b128_t* lds_b128_ptr;
#else
#define HAVE_ASYNC_LDS 0
#endif

// ---------------------------------------------------------------------------
// Kernel 1: bulk copy  key_past -> out[0:PER], value_past -> out[PER:2*PER].
// Each block handles one TILE; wave-private LDS slots => only per-wave
// ASYNCcnt fencing needed (no barriers).
// ---------------------------------------------------------------------------
__global__ __launch_bounds__(BLK) void kv_bulk_copy(const v4f* __restrict__ kp,
                                                    const v4f* __restrict__ vp,
                                                    v4f* __restrict__ out) {
    const unsigned tid = threadIdx.x;
#if HAVE_ASYNC_LDS
    __shared__ b128_t smem[TILE];
    for (size_t tile = (size_t)blockIdx.x * TILE; tile < TOTAL_T4;
         tile += (size_t)gridDim.x * TILE) {
        // drain this wave's prior async stores before overwriting its LDS slots
        __builtin_amdgcn_s_wait_asynccnt(0);
#pragma unroll
        for (int c = 0; c < CHUNK; ++c) {
            const size_t i  = tile + (size_t)c * BLK + tid;
            const v4f* src  = (i < PER_T4) ? (kp + i) : (vp + (i - PER_T4));
            __builtin_amdgcn_global_load_async_to_lds_b128(
                (glb_b128_ptr)(b128_t*)src,
                (lds_b128_ptr)&smem[c * BLK + tid], 0, CPOL_NT);
        }
        __builtin_amdgcn_s_wait_asynccnt(0);
#pragma unroll
        for (int c = 0; c < CHUNK; ++c) {
            const size_t i = tile + (size_t)c * BLK + tid;
            __builtin_amdgcn_global_store_async_from_lds_b128(
                (glb_b128_ptr)(b128_t*)(out + i),
                (lds_b128_ptr)&smem[c * BLK + tid], 0, CPOL_NT);
        }
    }
    __builtin_amdgcn_s_wait_asynccnt(0);
#else
    // Fallback: NT B128 copy through VGPRs (still fully streaming).
    for (size_t tile = (size_t)blockIdx.x * TILE; tile < TOTAL_T4;
         tile += (size_t)gridDim.x * TILE) {
#pragma unroll
        for (int c = 0; c < CHUNK; ++c) {
            const size_t i = tile + (size_t)c * BLK + tid;
            const v4f* src = (i < PER_T4) ? (kp + i) : (vp + (i - PER_T4));
            v4f v = __builtin_nontemporal_load(src);
            __builtin_nontemporal_store(v, out + i);
        }
    }
#endif
}

// ---------------------------------------------------------------------------
// Kernel 2: scatter xk/xv ([S,H,B,D] -> [B,H,past_len+S,D]) into out.
// One 16B packet per thread; rows are 512 B contiguous on both sides so all
// B128 accesses stay coalesced and control flow is wave-uniform.
// past_len read in-kernel (uniform scalar load) -> graph-capture safe.
// ---------------------------------------------------------------------------
__global__ __launch_bounds__(BLK) void kv_insert(const v4f* __restrict__ xk,
                                                 const v4f* __restrict__ xv,
                                                 const int* __restrict__ plen,
                                                 v4f* __restrict__ out) {
    const int past = *plen;  // uniform -> s_load
    const size_t i = (size_t)blockIdx.x * blockDim.x + threadIdx.x;  // [0, 2*X_T4)
    const size_t t = (i < X_T4) ? 0 : 1;
    const size_t r = i - t * X_T4;

    // xk/xv flat4 = ((s*8 + h)*8 + b)*32 + d4
    const unsigned d4 = (unsigned)(r & 31u);
    const unsigned b  = (unsigned)((r >> 5) & 7u);
    const unsigned h  = (unsigned)((r >> 8) & 7u);
    const unsigned s  = (unsigned)(r >> 11);  // 0..127

    const v4f* src = (t == 0) ? xk : xv;
    const v4f v = __builtin_nontemporal_load(src + r);

    // out flat4 = tensor*PER + ((b*8 + h)*4096 + past + s)*32 + d4
    const size_t dst = t * PER_T4 +
                       ((((size_t)b * N_KV_HEADS + h) * MAX_SEQ) + (size_t)past + s) *
                           (HEAD_DIM / 4) +
                       d4;
    __builtin_nontemporal_store(v, out + dst);
}

// ---------------------------------------------------------------------------
extern "C" void kernel_launch(void* const* d_in, const int* in_sizes, int n_in,
                              void* d_out, int out_size, void* d_ws, size_t ws_size,
                              hipStream_t stream) {
    const v4f* xk  = (const v4f*)d_in[0];
    const v4f* xv  = (const v4f*)d_in[1];
    const v4f* kp  = (const v4f*)d_in[2];
    const v4f* vp  = (const v4f*)d_in[3];
    const int* plen = (const int*)d_in[4];
    v4f* out       = (v4f*)d_out;

    const unsigned bulk_blocks   = (unsigned)(TOTAL_T4 / TILE);      // 16384
    const unsigned insert_blocks = (unsigned)(2 * X_T4 / BLK);       // 2048

    kv_bulk_copy<<<bulk_blocks, BLK, 0, stream>>>(kp, vp, out);
    kv_insert<<<insert_blocks, BLK, 0, stream>>>(xk, xv, plen, out);
}